// TripletDecoder_6339371729274
// MI455X (gfx1250) — compile-verified
//
#include <hip/hip_runtime.h>
#include <hip/hip_bf16.h>
#include <math.h>

// Problem constants (from reference)
#define MAXJ 64    // MAX_JETS (N)
#define BAT  128   // BATCH    (B)
#define HID  128   // HIDDEN   (d)
#define KTOT (HID * HID)  // 16384, flattened-W inner dim

typedef __attribute__((ext_vector_type(2))) float v2f;
typedef __attribute__((ext_vector_type(8))) float v8f;

// scale = 1 / (d*d*sqrt(d)), d = 128
__device__ __constant__ float kScaleConst = (float)(1.0 / (16384.0 * 11.313708498984760390));
#define K_SCALE ((float)(1.0 / (16384.0 * 11.313708498984760390)))

__device__ __forceinline__ v8f wmma_f32(v2f a, v2f b, v8f c) {
    // D = A(16x4,f32) * B(4x16,f32) + C(16x16,f32)  -> v_wmma_f32_16x16x4_f32
    return __builtin_amdgcn_wmma_f32_16x16x4_f32(
        /*neg_a=*/false, a, /*neg_b=*/false, b,
        /*c_mod=*/(short)0, c, /*reuse_a=*/false, /*reuse_b=*/false);
}

// ---------------------------------------------------------------------------
// K1: G[n][m] = sum_k Wf[n][k]*Wf[m][k]   (Wf = W viewed 128 x 16384)
//     + t[n] = trace(W[n])
// grid = 64 blocks (8x8 tiles of 16x16), 256 threads = 8 waves split K.
// ---------------------------------------------------------------------------
__global__ __launch_bounds__(256) void k_gram(const float* __restrict__ W,
                                              float* __restrict__ G,
                                              float* __restrict__ t) {
    const int tid  = threadIdx.x;
    const int lane = tid & 31;
    const int wv   = tid >> 5;
    const int bm   = blockIdx.x >> 3;
    const int bn   = blockIdx.x & 7;
    const int half = lane >> 4;
    const int r    = lane & 15;

    const float* rowA = W + (size_t)(bm * 16 + r) * KTOT;
    const float* rowB = W + (size_t)(bn * 16 + r) * KTOT;

    v8f c = {};
    int k = wv * 2048 + half * 2;           // each wave covers 2048 of K
    for (int i = 0; i < 512; ++i, k += 4) { // 512 wmma steps of K=4
        v2f a = *(const v2f*)(rowA + k);
        v2f b = *(const v2f*)(rowB + k);
        c = wmma_f32(a, b, c);
    }

    __shared__ float red[8][32][8];
#pragma unroll
    for (int j = 0; j < 8; ++j) red[wv][lane][j] = c[j];
    __syncthreads();

    if (wv == 0) {
#pragma unroll
        for (int j = 0; j < 8; ++j) {
            float s = 0.f;
#pragma unroll
            for (int w2 = 0; w2 < 8; ++w2) s += red[w2][lane][j];
            G[(bm * 16 + half * 8 + j) * HID + bn * 16 + r] = s;
        }
    }

    if (blockIdx.x == 0 && tid < HID) {   // trace
        float s = 0.f;
        const float* wr = W + (size_t)tid * KTOT;
        for (int i = 0; i < HID; ++i) s += wr[i * (HID + 1)];
        t[tid] = s;
    }
}

// ---------------------------------------------------------------------------
// K2: per batch b:  v = Xb*t ;  Y = Xb*G ;  P = Y*Xb^T ; masked max of
//     P[n,m]*v[l]*scale over valid distinct (n,m,l).
// grid = 128 blocks, 256 threads = 8 waves.
// ---------------------------------------------------------------------------
__global__ __launch_bounds__(256) void k_pv(const float* __restrict__ x,
                                            const float* __restrict__ seqmask,
                                            const float* __restrict__ G,
                                            const float* __restrict__ t,
                                            float* __restrict__ P,
                                            float* __restrict__ v,
                                            float* __restrict__ maxb) {
    const int b    = blockIdx.x;
    const int tid  = threadIdx.x;
    const int lane = tid & 31;
    const int wv   = tid >> 5;
    const int half = lane >> 4;
    const int r    = lane & 15;

    __shared__ float Ys[MAXJ * 132];   // Y staged, padded stride (bank-safe)
    __shared__ float Ps[MAXJ * 65];    // P staged for max pass
    __shared__ float vsh[MAXJ];
    __shared__ float sq[MAXJ];
    __shared__ float red[256];

    if (tid < MAXJ) {
        sq[tid] = seqmask[tid * BAT + b];            // sequence_mask[n][b][0]
        const float* xr = x + ((size_t)tid * BAT + b) * HID;
        float s = 0.f;
        for (int i = 0; i < HID; ++i) s += xr[i] * t[i];
        vsh[tid] = s;
        v[b * MAXJ + tid] = s;
    }

    // ---- GEMM1: Y(64x128) = Xb(64x128) * G(128x128); wave wv -> col tile cn=wv
    {
        const int cn = wv;
        v8f acc[4] = {};
        for (int k0 = 0; k0 < HID; k0 += 4) {
            const int kk = k0 + half * 2;
            v2f bf;
            bf[0] = G[(kk + 0) * HID + cn * 16 + r];
            bf[1] = G[(kk + 1) * HID + cn * 16 + r];
#pragma unroll
            for (int am = 0; am < 4; ++am) {
                const float* xr = x + ((size_t)(am * 16 + r) * BAT + b) * HID;
                v2f af = *(const v2f*)(xr + kk);
                acc[am] = wmma_f32(af, bf, acc[am]);
            }
        }
#pragma unroll
        for (int am = 0; am < 4; ++am)
#pragma unroll
            for (int j = 0; j < 8; ++j)
                Ys[(am * 16 + half * 8 + j) * 132 + cn * 16 + r] = acc[am][j];
    }
    __syncthreads();

    // ---- GEMM2: P(64x64) = Y(64x128) * Xb^T(128x64)
    {
        const int cn  = wv & 3;
        const int am0 = wv >> 2;
        v8f acc[2] = {};
        const float* xcol = x + ((size_t)(cn * 16 + r) * BAT + b) * HID;
        for (int k0 = 0; k0 < HID; k0 += 4) {
            const int kk = k0 + half * 2;
            v2f bf = *(const v2f*)(xcol + kk);
#pragma unroll
            for (int i = 0; i < 2; ++i) {
                const int am = am0 + i * 2;
                v2f af = *(const v2f*)(&Ys[(am * 16 + r) * 132 + kk]);
                acc[i] = wmma_f32(af, bf, acc[i]);
            }
        }
#pragma unroll
        for (int i = 0; i < 2; ++i) {
            const int am = am0 + i * 2;
#pragma unroll
            for (int j = 0; j < 8; ++j) {
                const int row = am * 16 + half * 8 + j;
                const int col = cn * 16 + r;
                const float val = acc[i][j];
                Ps[row * 65 + col] = val;
                P[((size_t)b * MAXJ + row) * MAXJ + col] = val;
            }
        }
    }
    __syncthreads();

    // ---- masked max of P[n,m]*scale*v[l]
    float mx = -3.402823466e38f;
    for (int nm = tid; nm < MAXJ * MAXJ; nm += 256) {
        const int n = nm >> 6, m = nm & 63;
        if (n == m || sq[n] <= 0.f || sq[m] <= 0.f) continue;
        const float a = Ps[n * 65 + m] * K_SCALE;
        for (int l = 0; l < MAXJ; ++l) {
            if (l == n || l == m || sq[l] <= 0.f) continue;
            mx = fmaxf(mx, a * vsh[l]);
        }
    }
    red[tid] = mx;
    __syncthreads();
    for (int s = 128; s > 0; s >>= 1) {
        if (tid < s) red[tid] = fmaxf(red[tid], red[tid + s]);
        __syncthreads();
    }
    if (tid == 0) maxb[b] = red[0];
}

// ---------------------------------------------------------------------------
// K3: partial sum-of-exp per (b, n) row.  grid = (64, 128), 256 threads.
// ---------------------------------------------------------------------------
__global__ __launch_bounds__(256) void k_sumexp(const float* __restrict__ P,
                                                const float* __restrict__ v,
                                                const float* __restrict__ seqmask,
                                                const float* __restrict__ maxb,
                                                float* __restrict__ partial) {
    const int n = blockIdx.x;
    const int b = blockIdx.y;
    const int tid = threadIdx.x;
    __shared__ float Pn[MAXJ], vsh[MAXJ], sq[MAXJ], red[256];
    if (tid < MAXJ) {
        Pn[tid]  = P[((size_t)b * MAXJ + n) * MAXJ + tid];
        vsh[tid] = v[b * MAXJ + tid];
        sq[tid]  = seqmask[tid * BAT + b];
    }
    __syncthreads();
    float s = 0.f;
    if (sq[n] > 0.f) {
        const float mx = maxb[b];
        for (int idx = tid; idx < MAXJ * MAXJ; idx += 256) {
            const int m = idx >> 6, l = idx & 63;
            if (m == n || l == n || l == m) continue;
            if (sq[m] <= 0.f || sq[l] <= 0.f) continue;
            s += __expf(Pn[m] * K_SCALE * vsh[l] - mx);
        }
    }
    red[tid] = s;
    __syncthreads();
    for (int st = 128; st > 0; st >>= 1) {
        if (tid < st) red[tid] += red[tid + st];
        __syncthreads();
    }
    if (tid == 0) partial[b * MAXJ + n] = red[0];
}

// ---------------------------------------------------------------------------
// K4: reduce 64 partials -> per-batch sum.  grid = 128 blocks, 64 threads.
// ---------------------------------------------------------------------------
__global__ __launch_bounds__(64) void k_redsum(const float* __restrict__ partial,
                                               float* __restrict__ sumb) {
    const int b = blockIdx.x;
    const int tid = threadIdx.x;
    __shared__ float red[64];
    red[tid] = partial[b * MAXJ + tid];
    __syncthreads();
    for (int st = 32; st > 0; st >>= 1) {
        if (tid < st) red[tid] += red[tid + st];
        __syncthreads();
    }
    if (tid == 0) sumb[b] = red[0];
}

// ---------------------------------------------------------------------------
// K5: streaming write of the masked softmax.  grid = (64, 128), 256 threads.
// ---------------------------------------------------------------------------
__global__ __launch_bounds__(256) void k_write(const float* __restrict__ P,
                                               const float* __restrict__ v,
                                               const float* __restrict__ seqmask,
                                               const float* __restrict__ maxb,
                                               const float* __restrict__ sumb,
                                               float* __restrict__ out) {
    const int n = blockIdx.x;
    const int b = blockIdx.y;
    const int tid = threadIdx.x;
    __shared__ float Pn[MAXJ], vsh[MAXJ], sq[MAXJ];
    if (tid < MAXJ) {
        Pn[tid]  = P[((size_t)b * MAXJ + n) * MAXJ + tid];
        vsh[tid] = v[b * MAXJ + tid];
        sq[tid]  = seqmask[tid * BAT + b];
    }
    __syncthreads();
    const float mx  = maxb[b];
    const float inv = 1.f / sumb[b];
    const bool rown = sq[n] > 0.f;
    float* o = out + ((size_t)b * MAXJ + n) * (MAXJ * MAXJ);
    for (int idx = tid; idx < MAXJ * MAXJ; idx += 256) {
        const int m = idx >> 6, l = idx & 63;
        const bool ok = rown && (sq[m] > 0.f) && (sq[l] > 0.f) &&
                        (n != m) && (n != l) && (m != l);
        o[idx] = ok ? __expf(Pn[m] * K_SCALE * vsh[l] - mx) * inv : 0.f;
    }
}

// ---------------------------------------------------------------------------
// Host launcher. Inputs (setup_inputs order):
//   d_in[0] = x              (64*128*128 f32)
//   d_in[1] = sequence_mask  (64*128*1  f32)
//   d_in[2] = W              (128^3     f32)
//   d_in[3] = padding_mask   (unused)
// Output: (128, 64, 64, 64) f32.
// Workspace layout (floats): G 16384 | t 128 | P 524288 | v 8192 |
//   maxb 128 | partial 8192 | sumb 128  (~2.23 MB total)
// ---------------------------------------------------------------------------
extern "C" void kernel_launch(void* const* d_in, const int* in_sizes, int n_in,
                              void* d_out, int out_size, void* d_ws, size_t ws_size,
                              hipStream_t stream) {
    const float* x   = (const float*)d_in[0];
    const float* sm  = (const float*)d_in[1];
    const float* W   = (const float*)d_in[2];
    (void)in_sizes; (void)n_in; (void)out_size; (void)ws_size;

    float* ws      = (float*)d_ws;
    float* G       = ws;                       // 16384
    float* t       = G + 16384;                // 128
    float* P       = t + 128;                  // 128*64*64 = 524288
    float* v       = P + 524288;               // 8192
    float* maxb    = v + 8192;                 // 128
    float* partial = maxb + 128;               // 8192
    float* sumb    = partial + 8192;           // 128

    k_gram  <<<64,              256, 0, stream>>>(W, G, t);
    k_pv    <<<128,             256, 0, stream>>>(x, sm, G, t, P, v, maxb);
    k_sumexp<<<dim3(64, 128),   256, 0, stream>>>(P, v, sm, maxb, partial);
    k_redsum<<<128,              64, 0, stream>>>(partial, sumb);
    k_write <<<dim3(64, 128),   256, 0, stream>>>(P, v, sm, maxb, sumb, (float*)d_out);
}